// Neigh_List_76484777607525
// MI455X (gfx1250) — compile-verified
//
#include <hip/hip_runtime.h>
#include <hip/hip_bf16.h>

typedef __attribute__((ext_vector_type(2))) float v2f;
typedef __attribute__((ext_vector_type(8))) float v8f;

#define CUTOFF   5.0f
// single-compare range test: 1e-6 < d2 < 25  ==>  |d2 - MID| < HALF
// MID = 12.5 + 1ulp, HALF = 12.5 - 1ulp, MID+HALF == 25.0 exactly (strict upper
// bound preserved); lower bound ~1.9e-6 only guards the exact self-pair d2==0.
#define DMID     12.500000953674316f
#define DHALF    12.499999046325684f
#define NCHUNK   64

// ---------------- wave helpers (wave32) ----------------
__device__ __forceinline__ unsigned ballot32(bool p) {
#if __has_builtin(__builtin_amdgcn_ballot_w32)
  return __builtin_amdgcn_ballot_w32(p);
#else
  return (unsigned)__ballot(p);
#endif
}
__device__ __forceinline__ int bperm_i(int srcLane, int val) {
  return __builtin_amdgcn_ds_bpermute(srcLane << 2, val);
}
__device__ __forceinline__ float bperm_f(int srcLane, float val) {
  return __int_as_float(bperm_i(srcLane, __float_as_int(val)));
}
__device__ __forceinline__ bool inrange(float d2) {
  return fabsf(d2 - DMID) < DHALF;   // 1 v_sub + 1 v_cmp (|.| is a src modifier)
}

// meta_i layout: 0:S 1:G 2:Gpad 3:jtiles 4:jtPerChunk 5..7:n2[xyz] 8:totalPairs
// meta_f layout: 0..2: maxcoor

// ---------------- K1: geometry prep (1 block, 256 thr) ----------------
__global__ __launch_bounds__(256)
void k_prep(const int* __restrict__ period,
            const float* __restrict__ coords,
            const float* __restrict__ cell,
            const float* __restrict__ mass,
            int N, int NPAD, int NRCAP,
            float* __restrict__ coord4,
            float* __restrict__ shiftVecs,
            int* __restrict__ meta_i,
            float* __restrict__ meta_f)
{
  const int t = threadIdx.x;
  const int NT = 256;
  __shared__ float s_inv[9];
  __shared__ float s_cell[9];
  __shared__ float s_ic0[3];
  __shared__ float s_com[3];
  __shared__ float s_min[3];
  __shared__ float s_red[4][256];
  __shared__ int   s_m[4];

  if (t < 9) s_cell[t] = cell[t];
  __syncthreads();
  if (t == 0) {
    float c00=s_cell[0],c01=s_cell[1],c02=s_cell[2];
    float c10=s_cell[3],c11=s_cell[4],c12=s_cell[5];
    float c20=s_cell[6],c21=s_cell[7],c22=s_cell[8];
    float det = c00*(c11*c22-c12*c21) - c01*(c10*c22-c12*c20) + c02*(c10*c21-c11*c20);
    float id = 1.0f/det;
    s_inv[0]=(c11*c22-c12*c21)*id; s_inv[1]=(c02*c21-c01*c22)*id; s_inv[2]=(c01*c12-c02*c11)*id;
    s_inv[3]=(c12*c20-c10*c22)*id; s_inv[4]=(c00*c22-c02*c20)*id; s_inv[5]=(c02*c10-c00*c12)*id;
    s_inv[6]=(c10*c21-c11*c20)*id; s_inv[7]=(c01*c20-c00*c21)*id; s_inv[8]=(c00*c11-c01*c10)*id;
    for (int j=0;j<3;j++)
      s_ic0[j] = coords[0]*s_inv[0*3+j] + coords[1]*s_inv[1*3+j] + coords[2]*s_inv[2*3+j];
  }
  __syncthreads();

  // wrap into cell relative to atom 0; accumulate mass-weighted sums
  float sx=0.f, sy=0.f, sz=0.f, sm=0.f;
  for (int a = t; a < N; a += NT) {
    float c0=coords[a*3+0], c1=coords[a*3+1], c2=coords[a*3+2];
    float ic[3], w[3], cc[3];
    for (int j=0;j<3;j++) ic[j] = c0*s_inv[0*3+j]+c1*s_inv[1*3+j]+c2*s_inv[2*3+j];
    for (int j=0;j<3;j++) w[j]  = ic[j] - rintf(ic[j]-s_ic0[j]);
    for (int j=0;j<3;j++) cc[j] = w[0]*s_cell[0*3+j]+w[1]*s_cell[1*3+j]+w[2]*s_cell[2*3+j];
    coord4[a*4+0]=cc[0]; coord4[a*4+1]=cc[1]; coord4[a*4+2]=cc[2];
    float mm = mass[a];
    sm += mm; sx += mm*cc[0]; sy += mm*cc[1]; sz += mm*cc[2];
  }
  s_red[0][t]=sx; s_red[1][t]=sy; s_red[2][t]=sz; s_red[3][t]=sm;
  __syncthreads();
  for (int s = 128; s > 0; s >>= 1) {
    if (t < s) { for (int k=0;k<4;k++) s_red[k][t] += s_red[k][t+s]; }
    __syncthreads();
  }
  if (t == 0) {
    float im = 1.0f / s_red[3][0];
    s_com[0]=s_red[0][0]*im; s_com[1]=s_red[1][0]*im; s_com[2]=s_red[2][0]*im;
  }
  __syncthreads();

  // subtract COM, reduce per-dim min
  float mn0=__builtin_inff(), mn1=__builtin_inff(), mn2=__builtin_inff();
  for (int a=t; a<N; a+=NT) {
    float v0 = coord4[a*4+0]-s_com[0]; coord4[a*4+0]=v0; mn0=fminf(mn0,v0);
    float v1 = coord4[a*4+1]-s_com[1]; coord4[a*4+1]=v1; mn1=fminf(mn1,v1);
    float v2 = coord4[a*4+2]-s_com[2]; coord4[a*4+2]=v2; mn2=fminf(mn2,v2);
  }
  s_red[0][t]=mn0; s_red[1][t]=mn1; s_red[2][t]=mn2;
  __syncthreads();
  for (int s=128;s>0;s>>=1){
    if (t<s) { for(int k=0;k<3;k++) s_red[k][t]=fminf(s_red[k][t],s_red[k][t+s]); }
    __syncthreads();
  }
  if (t==0) { for (int k=0;k<3;k++) s_min[k] = s_red[k][0] - CUTOFF - 1e-6f; }
  __syncthreads();

  // shift positive, store |r|^2, reduce per-dim max; poison pad rows
  float mx0=-__builtin_inff(), mx1=-__builtin_inff(), mx2=-__builtin_inff();
  for (int a=t; a<N; a+=NT) {
    float p0 = coord4[a*4+0]-s_min[0]; coord4[a*4+0]=p0; mx0=fmaxf(mx0,p0);
    float p1 = coord4[a*4+1]-s_min[1]; coord4[a*4+1]=p1; mx1=fmaxf(mx1,p1);
    float p2 = coord4[a*4+2]-s_min[2]; coord4[a*4+2]=p2; mx2=fmaxf(mx2,p2);
    coord4[a*4+3] = p0*p0 + p1*p1 + p2*p2;
  }
  for (int a=N+t; a<NPAD; a+=NT) {
    coord4[a*4+0]=3e9f; coord4[a*4+1]=0.f; coord4[a*4+2]=0.f; coord4[a*4+3]=9e18f;
  }
  s_red[0][t]=mx0; s_red[1][t]=mx1; s_red[2][t]=mx2;
  __syncthreads();
  for (int s=128;s>0;s>>=1){
    if (t<s) { for(int k=0;k<3;k++) s_red[k][t]=fmaxf(s_red[k][t],s_red[k][t+s]); }
    __syncthreads();
  }
  if (t==0) {
    for (int k=0;k<3;k++) meta_f[k] = s_red[k][0] + CUTOFF;
    int n2[3];
    for (int d=0; d<3; d++) {
      float cm = __builtin_inff();
      for (int i=0;i<3;i++){ float v=fabsf(s_cell[i*3+d]); if (v>1e-30f) cm=fminf(cm, CUTOFF/v); }
      int nr = (cm==__builtin_inff()) ? 0 : (int)ceilf(cm);
      nr *= period[d];
      if (nr > NRCAP) nr = NRCAP;
      if (nr < 0) nr = 0;
      n2[d] = 2*nr+1;
    }
    int S = n2[0]*n2[1]*n2[2];
    int G = S*N;
    int Gpad = (G + 15) & ~15;
    int jtiles = Gpad >> 4;
    int jtPer = (jtiles + NCHUNK - 1)/NCHUNK;
    meta_i[0]=S; meta_i[1]=G; meta_i[2]=Gpad; meta_i[3]=jtiles; meta_i[4]=jtPer;
    meta_i[5]=n2[0]; meta_i[6]=n2[1]; meta_i[7]=n2[2];
    s_m[0]=S; s_m[1]=n2[0]; s_m[2]=n2[1]; s_m[3]=n2[2];
  }
  __syncthreads();
  // shift vectors: meshgrid 'ij' flatten order (x-major, z-fastest), frac @ cell
  int S=s_m[0], n2y=s_m[2], n2z=s_m[3];
  int nrx=(s_m[1]-1)>>1, nry=(n2y-1)>>1, nrz=(n2z-1)>>1;
  for (int s=t; s<S; s+=NT) {
    int ix = s/(n2y*n2z); int rem = s - ix*(n2y*n2z);
    int iy = rem/n2z;     int iz  = rem - iy*n2z;
    float fx = (float)(ix-nrx), fy = (float)(iy-nry), fz = (float)(iz-nrz);
    for (int j=0;j<3;j++)
      shiftVecs[s*3+j] = fx*s_cell[0*3+j] + fy*s_cell[1*3+j] + fz*s_cell[2*3+j];
  }
}

// ---------------- K2: build image table (poison non-kept / pad) ----------------
__global__ __launch_bounds__(256)
void k_img(const float* __restrict__ coord4,
           const float* __restrict__ shiftVecs,
           const int* __restrict__ meta_i,
           const float* __restrict__ meta_f,
           int N,
           float* __restrict__ img4,
           unsigned* __restrict__ aux)
{
  int g = blockIdx.x*256 + threadIdx.x;
  int G = meta_i[1], Gpad = meta_i[2];
  if (g >= Gpad) return;
  bool ok = false;
  float px=0.f, py=0.f, pz=0.f;
  unsigned a = 0, s = 0;
  if (g < G) {
    s = (unsigned)g / (unsigned)N;
    a = (unsigned)g - s*(unsigned)N;
    px = coord4[a*4+0] + shiftVecs[s*3+0];
    py = coord4[a*4+1] + shiftVecs[s*3+1];
    pz = coord4[a*4+2] + shiftVecs[s*3+2];
    ok = (px > 0.f) && (px < meta_f[0]) &&
         (py > 0.f) && (py < meta_f[1]) &&
         (pz > 0.f) && (pz < meta_f[2]);
  }
  if (ok) {
    float w = px*px + py*py + pz*pz;
    img4[g*4+0] = -2.f*px; img4[g*4+1] = -2.f*py; img4[g*4+2] = -2.f*pz; img4[g*4+3] = w;
    aux[g] = (s << 16) | a;
  } else {
    img4[g*4+0] = -2e9f; img4[g*4+1]=0.f; img4[g*4+2]=0.f; img4[g*4+3]=1e18f;
    aux[g] = 0u;
  }
}

// d^2(m,n) as one 16x16x4 f32 WMMA:
//   A row m = (x, y, z, 1), B col n = (-2x', -2y', -2z', |r'|^2), C[m][n] = |r|^2

// ---------------- K3: WMMA count pass (lane-local arithmetic counting) ---------
__global__ __launch_bounds__(32)
void k_count(const float4* __restrict__ coord4,
             const float4* __restrict__ img4,
             const int* __restrict__ meta_i,
             int* __restrict__ cnt)
{
  const int lane  = threadIdx.x;
  const int bid   = blockIdx.x;
  const int itile = bid / NCHUNK;
  const int chunk = bid - itile*NCHUNK;
  const int jtiles = meta_i[3];
  const int jtPer  = meta_i[4];
  int jt0 = chunk * jtPer;
  int jt1 = jt0 + jtPer; if (jt1 > jtiles) jt1 = jtiles;
  const int i0 = itile << 4;
  const int m  = lane & 15;
  const int hl = lane >> 4;

  float4 cp = coord4[i0 + m];
  v2f A; A.x = hl ? cp.z : cp.x; A.y = hl ? 1.0f : cp.y;

  float creg[8];
  #pragma unroll
  for (int v=0; v<8; v++) creg[v] = bperm_f(v + hl*8, cp.w);

  int rc[8];
  #pragma unroll
  for (int v=0; v<8; v++) rc[v] = 0;

  // software-pipelined: issue next tile's load before processing current
  float4 bv = make_float4(0.f,0.f,0.f,0.f);
  if (jt0 < jt1) bv = img4[jt0*16 + m];
  for (int jt = jt0; jt < jt1; ++jt) {
    int jn = (jt + 1 < jt1) ? (jt + 1) : jt;
    float4 bnext = img4[jn*16 + m];
    if (jt + 16 < jt1) __builtin_prefetch(&img4[(jt+16)*16 + m], 0, 1);

    v2f B; B.x = hl ? bv.z : bv.x; B.y = hl ? bv.w : bv.y;
    v8f C;
    #pragma unroll
    for (int v=0; v<8; v++) C[v] = creg[v];
    v8f D = __builtin_amdgcn_wmma_f32_16x16x4_f32(
        false, A, false, B, (short)0, C, false, false);
    #pragma unroll
    for (int v=0; v<8; v++) rc[v] += inrange(D[v]) ? 1 : 0;

    bv = bnext;
  }
  // reduce across the 16 columns of this lane's half (xor masks stay in-half)
  #pragma unroll
  for (int v=0; v<8; v++) {
    int x = rc[v];
    x += __shfl_xor(x, 1);
    x += __shfl_xor(x, 2);
    x += __shfl_xor(x, 4);
    x += __shfl_xor(x, 8);
    rc[v] = x;
  }
  // lane (m<8) of half hl writes row (hl*8 + m): value rc[m]
  int out = rc[0];
  out = (m==1)?rc[1]:out; out = (m==2)?rc[2]:out; out = (m==3)?rc[3]:out;
  out = (m==4)?rc[4]:out; out = (m==5)?rc[5]:out; out = (m==6)?rc[6]:out;
  out = (m==7)?rc[7]:out;
  if (m < 8) cnt[(i0 + hl*8 + m) * NCHUNK + chunk] = out;
}

// ---------------- K4: exclusive scan over (center-major, chunk-minor) ----------
__global__ __launch_bounds__(1024)
void k_scan(const int* __restrict__ cnt, int* __restrict__ pairStart, int n,
            int* __restrict__ meta_i)
{
  const int t = threadIdx.x;
  const int NT = 1024;
  __shared__ int sh[1024];
  int per = (n + NT - 1)/NT;
  int b0 = t*per; if (b0 > n) b0 = n;
  int b1 = b0+per; if (b1 > n) b1 = n;
  int s = 0;
  for (int i=b0;i<b1;i++) s += cnt[i];
  sh[t]=s; __syncthreads();
  if (t==0) {
    int acc=0;
    for (int i=0;i<NT;i++){ int v=sh[i]; sh[i]=acc; acc+=v; }
    meta_i[8]=acc;
  }
  __syncthreads();
  int acc=sh[t];
  for (int i=b0;i<b1;i++){ pairStart[i]=acc; acc+=cnt[i]; }
}

// ---------------- K5: WMMA fill pass (uniform early-skip of empty tiles) ------
__global__ __launch_bounds__(32)
void k_fill(const float4* __restrict__ coord4,
            const float4* __restrict__ img4,
            const unsigned* __restrict__ aux,
            const float* __restrict__ shiftVecs,
            const int* __restrict__ meta_i,
            const int* __restrict__ pairStart,
            float* __restrict__ outC,
            float* __restrict__ outJ,
            float* __restrict__ outS,
            int P)
{
  const int lane  = threadIdx.x;
  const int bid   = blockIdx.x;
  const int itile = bid / NCHUNK;
  const int chunk = bid - itile*NCHUNK;
  const int jtiles = meta_i[3];
  const int jtPer  = meta_i[4];
  int jt0 = chunk * jtPer;
  int jt1 = jt0 + jtPer; if (jt1 > jtiles) jt1 = jtiles;
  const int i0 = itile << 4;
  const int m  = lane & 15;
  const int hl = lane >> 4;

  float4 cp = coord4[i0 + m];
  v2f A; A.x = hl ? cp.z : cp.x; A.y = hl ? 1.0f : cp.y;

  float creg[8];
  #pragma unroll
  for (int v=0; v<8; v++) creg[v] = bperm_f(v + hl*8, cp.w);

  int off = pairStart[(i0 + m) * NCHUNK + chunk];

  float4 bv = make_float4(0.f,0.f,0.f,0.f);
  if (jt0 < jt1) bv = img4[jt0*16 + m];
  for (int jt = jt0; jt < jt1; ++jt) {
    int jn = (jt + 1 < jt1) ? (jt + 1) : jt;
    float4 bnext = img4[jn*16 + m];
    if (jt + 16 < jt1) __builtin_prefetch(&img4[(jt+16)*16 + m], 0, 1);

    v2f B; B.x = hl ? bv.z : bv.x; B.y = hl ? bv.w : bv.y;
    v8f C;
    #pragma unroll
    for (int v=0; v<8; v++) C[v] = creg[v];
    v8f D = __builtin_amdgcn_wmma_f32_16x16x4_f32(
        false, A, false, B, (short)0, C, false, false);

    // lane-local predicate bitmask over the 8 accumulator rows
    unsigned lp = 0u;
    #pragma unroll
    for (int v=0; v<8; v++) lp |= (inrange(D[v]) ? 1u : 0u) << v;

    // uniform skip: >99% of tiles have no match at ~0.07% pair density
    if (ballot32(lp != 0u) != 0u) {
      #pragma unroll
      for (int v=0; v<8; v++) {
        bool pred = (lp >> v) & 1u;
        unsigned mask  = ballot32(pred);
        unsigned lowm  = mask & 0xffffu;   // row v,   cols 0..15
        unsigned highm = mask >> 16;       // row v+8, cols 0..15
        int row  = v + hl*8;
        int base = bperm_i(row, off);      // running offset lives in lane == row
        unsigned hm = hl ? highm : lowm;
        int rank = __popc(hm & ((1u << m) - 1u));
        if (pred) {
          int p = base + rank;
          if (p < P) {
            unsigned ax = aux[jt*16 + m];
            int aj = (int)(ax & 0xffffu);
            int sh = (int)(ax >> 16);
            outC[p] = (float)(i0 + row);
            outJ[p] = (float)aj;
            outS[p*3+0] = shiftVecs[sh*3+0];
            outS[p*3+1] = shiftVecs[sh*3+1];
            outS[p*3+2] = shiftVecs[sh*3+2];
          }
        }
        off += (lane == v    ) ? __popc(lowm)  : 0;
        off += (lane == v + 8) ? __popc(highm) : 0;
      }
    }
    bv = bnext;
  }
}

// ---------------- host ----------------
extern "C" void kernel_launch(void* const* d_in, const int* in_sizes, int n_in,
                              void* d_out, int out_size, void* d_ws, size_t ws_size,
                              hipStream_t stream) {
  const int*   period = (const int*)  d_in[0];
  const float* coords = (const float*)d_in[1];
  const float* cell   = (const float*)d_in[2];
  const float* mass   = (const float*)d_in[3];
  const int N      = in_sizes[1] / 3;
  const int NPAD   = (N + 15) & ~15;
  const int ITILES = NPAD >> 4;

  auto need = [&](int scap)->size_t {
    size_t gpadcap = (((size_t)scap*(size_t)N) + 15) & ~(size_t)15;
    size_t b = 256;                              // meta
    b += (size_t)NPAD*16;                        // coord4
    b += (((size_t)scap*12) + 15) & ~(size_t)15; // shiftVecs
    b += gpadcap*16;                             // img4
    b += gpadcap*4;                              // aux
    b += (size_t)NPAD*NCHUNK*4;                  // cnt
    b += (size_t)NPAD*NCHUNK*4;                  // pairStart
    return b + 256;
  };
  int NRCAP = 2;                                 // shifts cap 5^3 = 125
  if (need(125) > ws_size) { NRCAP = 1; if (need(27) > ws_size) NRCAP = 0; }
  int SCAP = 2*NRCAP+1; SCAP = SCAP*SCAP*SCAP;
  size_t GPADCAP = (((size_t)SCAP*(size_t)N) + 15) & ~(size_t)15;

  char* w = (char*)d_ws;
  int*      meta_i    = (int*)w;                 // 16 ints
  float*    meta_f    = (float*)(w + 64);
  size_t off = 256;
  float*    coord4    = (float*)(w + off); off += (size_t)NPAD*16;
  float*    shiftVecs = (float*)(w + off); off += (((size_t)SCAP*12) + 15) & ~(size_t)15;
  float*    img4      = (float*)(w + off); off += GPADCAP*16;
  unsigned* aux       = (unsigned*)(w + off); off += GPADCAP*4;
  int*      cnt       = (int*)(w + off); off += (size_t)NPAD*NCHUNK*4;
  int*      pairStart = (int*)(w + off); off += (size_t)NPAD*NCHUNK*4;

  const int P = out_size / 5;                    // 2P indices + 3P shift floats
  float* outC = (float*)d_out;
  float* outJ = outC + (size_t)P;
  float* outS = outC + 2*(size_t)P;

  k_prep<<<1, 256, 0, stream>>>(period, coords, cell, mass, N, NPAD, NRCAP,
                                coord4, shiftVecs, meta_i, meta_f);
  int gi = (int)((GPADCAP + 255)/256);
  k_img<<<gi, 256, 0, stream>>>(coord4, shiftVecs, meta_i, meta_f, N, img4, aux);
  k_count<<<ITILES*NCHUNK, 32, 0, stream>>>((const float4*)coord4, (const float4*)img4,
                                            meta_i, cnt);
  k_scan<<<1, 1024, 0, stream>>>(cnt, pairStart, NPAD*NCHUNK, meta_i);
  k_fill<<<ITILES*NCHUNK, 32, 0, stream>>>((const float4*)coord4, (const float4*)img4,
                                           aux, shiftVecs, meta_i, pairStart,
                                           outC, outJ, outS, P);
}